// Cal_adj_matrix_63067299775200
// MI455X (gfx1250) — compile-verified
//
#include <hip/hip_runtime.h>

typedef __attribute__((ext_vector_type(2))) float v2f;
typedef __attribute__((ext_vector_type(8))) float v8f;

#define B_DIM 32
#define C_DIM 256
#define H_DIM 96
#define W_DIM 96
#define M_DIM (H_DIM * W_DIM)   // 9216
#define N_OUT 50
#define N_PAD 64
#define ROWS_PER_WG 128          // 4 waves x 32 rows

// ---------------------------------------------------------------------------
// Kernel 1: pyramid pool. One block per (b,c) image (8192 blocks, 192 thr).
// Produces pool[b*c][64] in workspace, padded with zeros for n in [50,64).
// ---------------------------------------------------------------------------
__global__ __launch_bounds__(192) void pool_kernel(const float* __restrict__ x,
                                                   float* __restrict__ pool) {
  __shared__ float colsum[576];   // (y, bx) : sum of 16 consecutive x
  __shared__ float bins[36];      // 6x6 grid of 16x16 bin sums
  const int bc = blockIdx.x;
  const int t  = threadIdx.x;
  const float* img = x + (size_t)bc * M_DIM;

  // Stage A: 576 row-segment sums (96 rows x 6 column blocks of 16)
  #pragma unroll
  for (int it = 0; it < 3; ++it) {
    int item = t + 192 * it;
    if (item < 576) {
      int y = item / 6, bx = item % 6;
      const float* p = img + y * W_DIM + bx * 16;
      float s = 0.f;
      #pragma unroll
      for (int j = 0; j < 16; ++j) s += p[j];
      colsum[item] = s;
    }
  }
  __syncthreads();

  // Stage B: 6x6 bin sums (16 rows each)
  if (t < 36) {
    int by = t / 6, bx = t % 6;
    float s = 0.f;
    #pragma unroll
    for (int j = 0; j < 16; ++j) s += colsum[(by * 16 + j) * 6 + bx];
    bins[t] = s;
  }
  __syncthreads();

  // Stage C: derive all 50 pool means from the 36 bins; pad to 64 with zeros.
  if (t < N_PAD) {
    float v = 0.f;
    if (t == 0) {                       // s=1 : mean of whole image
      float s = 0.f;
      #pragma unroll
      for (int i = 0; i < 36; ++i) s += bins[i];
      v = s * (1.0f / 9216.0f);
    } else if (t < 5) {                 // s=2 : 3x3 groups of bins (48x48 px)
      int i = t - 1;
      int by = (i >> 1) * 3, bx = (i & 1) * 3;
      float s = 0.f;
      #pragma unroll
      for (int dy = 0; dy < 3; ++dy)
        #pragma unroll
        for (int dx = 0; dx < 3; ++dx) s += bins[(by + dy) * 6 + (bx + dx)];
      v = s * (1.0f / 2304.0f);
    } else if (t < 14) {                // s=3 : 2x2 groups of bins (32x32 px)
      int i = t - 5;
      int by = (i / 3) * 2, bx = (i % 3) * 2;
      v = (bins[by * 6 + bx] + bins[by * 6 + bx + 1] +
           bins[(by + 1) * 6 + bx] + bins[(by + 1) * 6 + bx + 1]) *
          (1.0f / 1024.0f);
    } else if (t < 50) {                // s=6 : the bins themselves (16x16 px)
      v = bins[t - 14] * (1.0f / 256.0f);
    }
    pool[(size_t)bc * N_PAD + t] = v;
  }
}

// ---------------------------------------------------------------------------
// Kernel 2: sim = feat^T x pool (per batch GEMM M=9216,N=64(pad),K=256) via
// V_WMMA_F32_16X16X4_F32, then fused relu / rowsum / scale / store.
// Grid: (72 row-blocks of 128, 32 batches); block = 128 threads = 4 waves.
// Each wave computes a 32x64 tile (2 M-tiles x 4 N-tiles, 8 accumulators):
// 8 WMMA per 12 loads per k-step.
// ---------------------------------------------------------------------------
__global__ __launch_bounds__(128) void adj_kernel(const float* __restrict__ x,
                                                  const float* __restrict__ pool,
                                                  float* __restrict__ out) {
  __shared__ float simLds[ROWS_PER_WG * N_PAD];  // 32 KB staging for finalize

  const int batch = blockIdx.y;
  const int mblk  = blockIdx.x;             // 0..71
  const int wave  = threadIdx.x >> 5;
  const int lane  = threadIdx.x & 31;
  const int half  = lane >> 4;              // K split across half-waves
  const int l16   = lane & 15;              // M (for A) / N (for B,D)

  const float* xb = x    + (size_t)batch * C_DIM * M_DIM;
  const float* pb = pool + (size_t)batch * C_DIM * N_PAD;
  const int m0 = mblk * ROWS_PER_WG + wave * 32;   // first of 32 rows

  v8f acc00 = {}, acc01 = {}, acc02 = {}, acc03 = {};   // rows m0 .. m0+15
  v8f acc10 = {}, acc11 = {}, acc12 = {}, acc13 = {};   // rows m0+16 .. m0+31

  // A-matrix 16x4 f32 layout: lanes 0-15 hold M=0..15 with VGPR0=K0,VGPR1=K1;
  // lanes 16-31 hold M=0..15 with VGPR0=K2,VGPR1=K3.  feat[c][m] = xb[c*M+m].
  const float* aBase0 = xb + m0 + l16;
  const float* aBase1 = aBase0 + 16;

  for (int k0 = 0; k0 < C_DIM; k0 += 4) {
    const int ka = k0 + 2 * half;
    v2f a0, a1;
    a0.x = aBase0[(size_t)ka * M_DIM];
    a0.y = aBase0[(size_t)(ka + 1) * M_DIM];
    a1.x = aBase1[(size_t)ka * M_DIM];
    a1.y = aBase1[(size_t)(ka + 1) * M_DIM];

    // B-matrix 4x16 f32: lanes 0-15 hold N with K0/K1 in the VGPR pair,
    // lanes 16-31 hold N with K2/K3.  B[k][n] = pb[k*64 + n].
    const float* bp = pb + (size_t)ka * N_PAD + l16;
    v2f b0; b0.x = bp[0];  b0.y = bp[N_PAD];
    v2f b1; b1.x = bp[16]; b1.y = bp[N_PAD + 16];
    v2f b2; b2.x = bp[32]; b2.y = bp[N_PAD + 32];
    v2f b3; b3.x = bp[48]; b3.y = bp[N_PAD + 48];

    acc00 = __builtin_amdgcn_wmma_f32_16x16x4_f32(false, a0, false, b0, (short)0, acc00, false, false);
    acc01 = __builtin_amdgcn_wmma_f32_16x16x4_f32(false, a0, false, b1, (short)0, acc01, false, false);
    acc02 = __builtin_amdgcn_wmma_f32_16x16x4_f32(false, a0, false, b2, (short)0, acc02, false, false);
    acc03 = __builtin_amdgcn_wmma_f32_16x16x4_f32(false, a0, false, b3, (short)0, acc03, false, false);
    acc10 = __builtin_amdgcn_wmma_f32_16x16x4_f32(false, a1, false, b0, (short)0, acc10, false, false);
    acc11 = __builtin_amdgcn_wmma_f32_16x16x4_f32(false, a1, false, b1, (short)0, acc11, false, false);
    acc12 = __builtin_amdgcn_wmma_f32_16x16x4_f32(false, a1, false, b2, (short)0, acc12, false, false);
    acc13 = __builtin_amdgcn_wmma_f32_16x16x4_f32(false, a1, false, b3, (short)0, acc13, false, false);
  }

  // sim = gemm / (b*c*h*w)
  const float invN = 1.0f / 75497472.0f;   // 32*256*96*96

  // D layout: VGPR i holds M=i (lanes 0-15) and M=8+i (lanes 16-31), N=l16.
  #pragma unroll
  for (int i = 0; i < 8; ++i) {
    int row0 = wave * 32 + i + 8 * half;
    int row1 = row0 + 16;
    simLds[row0 * N_PAD +  0 + l16] = acc00[i] * invN;
    simLds[row0 * N_PAD + 16 + l16] = acc01[i] * invN;
    simLds[row0 * N_PAD + 32 + l16] = acc02[i] * invN;
    simLds[row0 * N_PAD + 48 + l16] = acc03[i] * invN;
    simLds[row1 * N_PAD +  0 + l16] = acc10[i] * invN;
    simLds[row1 * N_PAD + 16 + l16] = acc11[i] * invN;
    simLds[row1 * N_PAD + 32 + l16] = acc12[i] * invN;
    simLds[row1 * N_PAD + 48 + l16] = acc13[i] * invN;
  }
  __syncthreads();

  // Finalize: relu, rowsum over the real 50 columns, scale, store.
  {
    const int r = threadIdx.x;               // one row per thread (128 rows)
    const float* srow = &simLds[r * N_PAD];
    float total = 0.f;
    #pragma unroll
    for (int n = 0; n < N_OUT; ++n) {
      float v = srow[n];
      total += (v > 0.f ? v : 0.f);
    }
    const float scale = 1.0f / (total * total + 1e-6f);
    float* orow = out + ((size_t)batch * M_DIM + (size_t)mblk * ROWS_PER_WG + r) * N_OUT;
    #pragma unroll
    for (int n = 0; n < N_OUT; ++n) {
      float v = srow[n];
      v = (v > 0.f ? v : 0.f);
      orow[n] = v * scale;
    }
  }
}

// ---------------------------------------------------------------------------
extern "C" void kernel_launch(void* const* d_in, const int* in_sizes, int n_in,
                              void* d_out, int out_size, void* d_ws, size_t ws_size,
                              hipStream_t stream) {
  const float* x  = (const float*)d_in[0];
  float* out      = (float*)d_out;
  float* poolbuf  = (float*)d_ws;   // needs 32*256*64*4 = 2 MB

  pool_kernel<<<dim3(B_DIM * C_DIM), 192, 0, stream>>>(x, poolbuf);
  adj_kernel<<<dim3(M_DIM / ROWS_PER_WG, B_DIM), 128, 0, stream>>>(x, poolbuf, out);
}